// DMIL_15058155340600
// MI455X (gfx1250) — compile-verified
//
#include <hip/hip_runtime.h>
#include <hip/hip_bf16.h>
#include <math.h>

typedef float v2f __attribute__((ext_vector_type(2)));
typedef float v8f __attribute__((ext_vector_type(8)));

#define FG_THRESH 0.5f
#define BG_THRESH 0.1f
#define PCL_EPS   1e-9f
#define NSTREAM   4
#define MAXC      32
#define MAXG      (MAXC * NSTREAM)
#define NSLICE    64

// Output layout (floats): [0]=loss, [1 .. 1+4G)=gt_boxes, [1+4G .. 1+5G)=gt_scores,
//                         [1+5G .. 1+6G)=valid (0/1)
// Workspace (floats): [0..G)=sums, [G..2G)=counts, [2G]=neg accumulator,
//                     [512..512+4*C*NSLICE)=partial argmax values,
//                     then 4*C*NSLICE ints = partial argmax indices

// -------- Kernel 1 (generic): per (stream,class) argmax over R --------------
__global__ void topk_argmax_kernel(const float* __restrict__ rois,
                                   const float* __restrict__ c0,
                                   const float* __restrict__ c1,
                                   const float* __restrict__ c2,
                                   const float* __restrict__ c3,
                                   float* __restrict__ out, int R, int C) {
  int s = blockIdx.x / C;
  int c = blockIdx.x % C;
  const float* cls = (s == 0) ? c0 : (s == 1) ? c1 : (s == 2) ? c2 : c3;
  __shared__ float sv[256];
  __shared__ int   si[256];
  int tid = threadIdx.x;
  float bv = -INFINITY; int bi = 0x7fffffff;
  for (int r = tid; r < R; r += 256) {
    float v = cls[(size_t)r * C + c];
    if (v > bv) { bv = v; bi = r; }
  }
  sv[tid] = bv; si[tid] = bi;
  __syncthreads();
  for (int off = 128; off > 0; off >>= 1) {
    if (tid < off) {
      float ov = sv[tid + off]; int oi = si[tid + off];
      if (ov > sv[tid] || (ov == sv[tid] && oi < si[tid])) { sv[tid] = ov; si[tid] = oi; }
    }
    __syncthreads();
  }
  if (tid == 0) {
    int idx = si[0];
    int g = c * NSTREAM + s;
    int G = C * NSTREAM;
    const float* bb = rois + (size_t)idx * 4;
    out[1 + g * 4 + 0] = bb[0];
    out[1 + g * 4 + 1] = bb[1];
    out[1 + g * 4 + 2] = bb[2];
    out[1 + g * 4 + 3] = bb[3];
    out[1 + 4 * G + g] = sv[0];
  }
}

// -------- Kernel 1a (specialized): coalesced row-scan, sliced ---------------
template <int CT>
__global__ void __launch_bounds__(256) topk_slice_kernel(
    const float* __restrict__ c0, const float* __restrict__ c1,
    const float* __restrict__ c2, const float* __restrict__ c3,
    float* __restrict__ pv, int* __restrict__ pi, int R) {
  int s = blockIdx.x / NSLICE;
  int slice = blockIdx.x % NSLICE;
  const float* cls = (s == 0) ? c0 : (s == 1) ? c1 : (s == 2) ? c2 : c3;
  int rowsPer = (R + NSLICE - 1) / NSLICE;
  int r0 = slice * rowsPer;
  int r1 = min(R, r0 + rowsPer);

  float bv[CT]; int bi[CT];
#pragma unroll
  for (int c = 0; c < CT; ++c) { bv[c] = -INFINITY; bi[c] = 0x7fffffff; }

  for (int r = r0 + (int)threadIdx.x; r < r1; r += 256) {
    const float4* row = (const float4*)(cls + (size_t)r * CT);  // CT*4 % 16 == 0
#pragma unroll
    for (int q = 0; q < CT / 4; ++q) {
      float4 v4 = row[q];
      if (v4.x > bv[4 * q + 0]) { bv[4 * q + 0] = v4.x; bi[4 * q + 0] = r; }
      if (v4.y > bv[4 * q + 1]) { bv[4 * q + 1] = v4.y; bi[4 * q + 1] = r; }
      if (v4.z > bv[4 * q + 2]) { bv[4 * q + 2] = v4.z; bi[4 * q + 2] = r; }
      if (v4.w > bv[4 * q + 3]) { bv[4 * q + 3] = v4.w; bi[4 * q + 3] = r; }
    }
  }

  __shared__ float sv[256];
  __shared__ int   si[256];
  int tid = threadIdx.x;
#pragma unroll
  for (int c = 0; c < CT; ++c) {
    sv[tid] = bv[c]; si[tid] = bi[c];
    __syncthreads();
    for (int off = 128; off > 0; off >>= 1) {
      if (tid < off) {
        float ov = sv[tid + off]; int oi = si[tid + off];
        if (ov > sv[tid] || (ov == sv[tid] && oi < si[tid])) { sv[tid] = ov; si[tid] = oi; }
      }
      __syncthreads();
    }
    if (tid == 0) {
      pv[(s * CT + c) * NSLICE + slice] = sv[0];
      pi[(s * CT + c) * NSLICE + slice] = si[0];
    }
    __syncthreads();
  }
}

// -------- Kernel 1b: deterministic slice reduction --------------------------
__global__ void topk_reduce_kernel(const float* __restrict__ rois,
                                   const float* __restrict__ pv,
                                   const int* __restrict__ pi,
                                   float* __restrict__ out, int C) {
  int t = threadIdx.x;
  int total = NSTREAM * C;
  if (t >= total) return;
  int s = t / C, c = t % C;
  float bv = -INFINITY; int bi = 0x7fffffff;
  for (int k = 0; k < NSLICE; ++k) {           // ascending slices: first-max wins
    float v = pv[(s * C + c) * NSLICE + k];
    int   i = pi[(s * C + c) * NSLICE + k];
    if (v > bv || (v == bv && i < bi)) { bv = v; bi = i; }
  }
  int g = c * NSTREAM + s;
  int G = C * NSTREAM;
  const float* bb = rois + (size_t)bi * 4;
  out[1 + g * 4 + 0] = bb[0];
  out[1 + g * 4 + 1] = bb[1];
  out[1 + g * 4 + 2] = bb[2];
  out[1 + g * 4 + 3] = bb[3];
  out[1 + 4 * G + g] = bv;
}

// -------- Kernel 2: per-class instance selector + zero accumulators ---------
__global__ void select_kernel(const int* __restrict__ im_labels,
                              float* __restrict__ out,
                              float* __restrict__ ws, int C) {
  int tid = threadIdx.x;
  int G = C * NSTREAM;
  for (int i = tid; i < 2 * G + 1; i += blockDim.x) ws[i] = 0.0f;
  if (tid < C) {
    bool present = im_labels[tid] > 0;
    float bx[NSTREAM][4], area[NSTREAM];
    for (int s = 0; s < NSTREAM; ++s) {
      int g = tid * NSTREAM + s;
      for (int k = 0; k < 4; ++k) bx[s][k] = out[1 + g * 4 + k];
      area[s] = (bx[s][2] - bx[s][0] + 1.0f) * (bx[s][3] - bx[s][1] + 1.0f);
    }
    bool active[NSTREAM], selected[NSTREAM];
    for (int s = 0; s < NSTREAM; ++s) { active[s] = present; selected[s] = false; }
    for (int it = 0; it < NSTREAM; ++it) {
      float ba = -INFINITY; int m = 0;
      for (int i = 0; i < NSTREAM; ++i) {
        float a = active[i] ? area[i] : -INFINITY;
        if (a > ba) { ba = a; m = i; }
      }
      float bm0 = bx[m][0], bm1 = bx[m][1], bm2 = bx[m][2], bm3 = bx[m][3];
      float am  = (bm2 - bm0 + 1.0f) * (bm3 - bm1 + 1.0f);
      bool nact[NSTREAM];
      for (int i = 0; i < NSTREAM; ++i) {
        float x1 = fmaxf(bx[i][0], bm0), y1 = fmaxf(bx[i][1], bm1);
        float x2 = fminf(bx[i][2], bm2), y2 = fminf(bx[i][3], bm3);
        float iw = fmaxf(x2 - x1 + 1.0f, 0.0f), ih = fmaxf(y2 - y1 + 1.0f, 0.0f);
        float inter = iw * ih;
        float iou = inter / (area[i] + am - inter);
        bool enclosed = (bx[i][0] > bm0) && (bx[i][1] > bm1) &&
                        (bx[i][2] < bm2) && (bx[i][3] < bm3);
        bool pick = active[i] && (i == m);
        bool keep = active[i] && (iou < 0.1f) && (!enclosed);
        selected[i] = selected[i] || pick || keep;
        nact[i] = active[i] && !((iou > 0.3f) || enclosed || keep);
      }
      for (int i = 0; i < NSTREAM; ++i) active[i] = nact[i];
    }
    for (int s = 0; s < NSTREAM; ++s)
      out[1 + 5 * G + tid * NSTREAM + s] = selected[s] ? 1.0f : 0.0f;
  }
}

// -------- Kernel 3: main PCL loss pass over all rois ------------------------
__device__ __forceinline__ float clamp_p(float v) {
  return fminf(fmaxf(v, PCL_EPS), 1.0f - PCL_EPS);
}

template <int CT>
__device__ __forceinline__ float roi_term(int r, int R, int Crt,
    const float* __restrict__ rois, const float* __restrict__ pnew,
    const float4* sGtb4, const float* sGts, const float* sValM, const float* sArea,
    float* sSum, float* sCnt) {
  const int C = CT ? CT : Crt;
  const int G = C * NSTREAM;
  if (r >= R) return 0.0f;
  float4 bb = ((const float4*)rois)[r];
  float ar = (bb.z - bb.x + 1.0f) * (bb.w - bb.y + 1.0f);
  // argmax of inter/den via cross-multiplication (den > 0 always);
  // invalid gts contribute ov == 0 (outcome-equivalent to reference's -1 mask).
  float bInter = -1.0f, bDen = 1.0f;
  int gi = 0;
#pragma unroll 8
  for (int g = 0; g < G; ++g) {
    float4 gt = sGtb4[g];
    float ix1 = fmaxf(bb.x, gt.x), iy1 = fmaxf(bb.y, gt.y);
    float ix2 = fminf(bb.z, gt.z), iy2 = fminf(bb.w, gt.w);
    float iw = fmaxf(ix2 - ix1 + 1.0f, 0.0f);
    float ih = fmaxf(iy2 - iy1 + 1.0f, 0.0f);
    float inter = iw * ih;
    float den = ar + sArea[g] - inter;
    float interM = inter * sValM[g];
    bool better = (interM * bDen) > (bInter * den);
    gi     = better ? g      : gi;
    bInter = better ? interM : bInter;
    bDen   = better ? den    : bDen;
  }
  float maxov = bInter / bDen;                 // one IEEE div per roi
  float w = (maxov < BG_THRESH) ? 0.0f : sGts[gi];
  if (maxov < FG_THRESH) {                     // background roi
    if (w != 0.0f) {
      float p0 = clamp_p(pnew[(size_t)r * (C + 1)]);
      return logf(p0) * w;                     // negated in finalize
    }
    return 0.0f;
  } else {                                     // foreground: segment accumulate
    int lab = gi / NSTREAM + 1;
    float pa = clamp_p(pnew[(size_t)r * (C + 1) + lab]);
    atomicAdd(&sSum[gi], pa);
    atomicAdd(&sCnt[gi], 1.0f);
    return 0.0f;
  }
}

template <int CT>
__global__ void __launch_bounds__(256) pcl_main_kernel(
    const float* __restrict__ rois, const float* __restrict__ pnew,
    float* __restrict__ out, float* __restrict__ ws, int R, int Crt) {
  const int C = CT ? CT : Crt;
  const int G = C * NSTREAM;
  __shared__ float4 sGtb4[MAXG];
  __shared__ float sGts[MAXG];
  __shared__ float sValM[MAXG];
  __shared__ float sArea[MAXG];
  __shared__ float sSum[MAXG];
  __shared__ float sCnt[MAXG];
  __shared__ float sNeg;
  int tid = threadIdx.x;
  for (int i = tid; i < G; i += blockDim.x) {
    float4 b;
    b.x = out[1 + i * 4 + 0]; b.y = out[1 + i * 4 + 1];
    b.z = out[1 + i * 4 + 2]; b.w = out[1 + i * 4 + 3];
    sGtb4[i] = b;
    sArea[i] = (b.z - b.x + 1.0f) * (b.w - b.y + 1.0f);
    sGts[i]  = out[1 + 4 * G + i];
    sValM[i] = (out[1 + 5 * G + i] > 0.5f) ? 1.0f : 0.0f;
    sSum[i] = 0.0f; sCnt[i] = 0.0f;
  }
  if (tid == 0) sNeg = 0.0f;
  __syncthreads();

  int lane = tid & 31;
  int wavesPerBlock = blockDim.x >> 5;
  int wave = blockIdx.x * wavesPerBlock + (tid >> 5);
  int nWaves = gridDim.x * wavesPerBlock;
  int nChunks = (R + 63) >> 6;                 // 64 rois per wave chunk

  // neg-term reduction through the matrix engine:
  // D = A x ones(4x16) + C, chained; lanes {0,16} jointly hold all 16 row sums.
  v8f acc = {0.f, 0.f, 0.f, 0.f, 0.f, 0.f, 0.f, 0.f};
  v2f ones2; ones2[0] = 1.0f; ones2[1] = 1.0f;

  for (int ch = wave; ch < nChunks; ch += nWaves) {
    int base = ch << 6;
    int nb = ch + nWaves;
    if (nb < nChunks) {                        // global_prefetch_b8 next chunk
      int pr = (nb << 6) + lane;
      __builtin_prefetch(rois + (size_t)pr * 4, 0, 1);
      __builtin_prefetch(rois + (size_t)(pr + 32) * 4, 0, 1);
    }
    float t0 = roi_term<CT>(base + lane,      R, Crt, rois, pnew,
                            sGtb4, sGts, sValM, sArea, sSum, sCnt);
    float t1 = roi_term<CT>(base + 32 + lane, R, Crt, rois, pnew,
                            sGtb4, sGts, sValM, sArea, sSum, sCnt);
    v2f a; a[0] = t0; a[1] = t1;
    acc = __builtin_amdgcn_wmma_f32_16x16x4_f32(
        /*neg_a=*/false, a, /*neg_b=*/false, ones2,
        /*c_mod=*/(short)0, acc, /*reuse_a=*/false, /*reuse_b=*/false);
  }
  float partial = acc[0] + acc[1] + acc[2] + acc[3] +
                  acc[4] + acc[5] + acc[6] + acc[7];
  if (lane == 0 || lane == 16) atomicAdd(&sNeg, partial);
  __syncthreads();

  for (int i = tid; i < G; i += blockDim.x) {
    if (sSum[i] != 0.0f) atomicAdd(&ws[i], sSum[i]);
    if (sCnt[i] != 0.0f) atomicAdd(&ws[G + i], sCnt[i]);
  }
  if (tid == 0 && sNeg != 0.0f) atomicAdd(&ws[2 * G], sNeg);
}

// -------- Kernel 4: finalize loss -------------------------------------------
__global__ void finalize_kernel(const float* __restrict__ ws,
                                float* __restrict__ out, int R, int C) {
  int G = C * NSTREAM;
  __shared__ float red[128];
  int tid = threadIdx.x;
  float acc = 0.0f;
  for (int g = tid; g < G; g += blockDim.x) {
    float cnt = ws[G + g];
    float vf  = out[1 + 5 * G + g];
    if (vf > 0.5f && cnt > 0.0f) {
      float mean = ws[g] / fmaxf(cnt, 1.0f);
      acc += logf(mean) * cnt * out[1 + 4 * G + g];
    }
  }
  red[tid] = acc;
  __syncthreads();
  for (int off = 64; off > 0; off >>= 1) {
    if (tid < off) red[tid] += red[tid + off];
    __syncthreads();
  }
  if (tid == 0) {
    float denom = (float)(R > 800 ? R : 800);
    out[0] = -(red[0] + ws[2 * G]) / denom;    // (pos + neg) / max(800, R)
  }
}

extern "C" void kernel_launch(void* const* d_in, const int* in_sizes, int n_in,
                              void* d_out, int out_size, void* d_ws, size_t ws_size,
                              hipStream_t stream) {
  const float* rois = (const float*)d_in[0];
  const float* c0   = (const float*)d_in[1];
  const float* c1   = (const float*)d_in[2];
  const float* c2   = (const float*)d_in[3];
  const float* c3   = (const float*)d_in[4];
  const float* pnew = (const float*)d_in[5];
  const int*   iml  = (const int*)d_in[6];
  float* out = (float*)d_out;
  float* ws  = (float*)d_ws;

  int R = in_sizes[0] / 4;
  int C = in_sizes[6];                 // 20 in the reference setup

  if (C == 20 && ws_size >= (size_t)(512 + 8 * 20 * NSLICE + 64) * sizeof(float)) {
    float* pv = ws + 512;
    int*   pi = (int*)(ws + 512 + NSTREAM * 20 * NSLICE);
    topk_slice_kernel<20><<<NSTREAM * NSLICE, 256, 0, stream>>>(c0, c1, c2, c3, pv, pi, R);
    topk_reduce_kernel<<<1, 128, 0, stream>>>(rois, pv, pi, out, C);
  } else {
    topk_argmax_kernel<<<NSTREAM * C, 256, 0, stream>>>(rois, c0, c1, c2, c3, out, R, C);
  }

  select_kernel<<<1, 64, 0, stream>>>(iml, out, ws, C);

  int nChunks = (R + 63) / 64;
  int blocks = (nChunks + 7) / 8;      // 8 waves per 256-thread block
  if (blocks > 256) blocks = 256;
  if (blocks < 1) blocks = 1;
  if (C == 20) {
    pcl_main_kernel<20><<<blocks, 256, 0, stream>>>(rois, pnew, out, ws, R, C);
  } else {
    pcl_main_kernel<0><<<blocks, 256, 0, stream>>>(rois, pnew, out, ws, R, C);
  }

  finalize_kernel<<<1, 128, 0, stream>>>(ws, out, R, C);
}